// CAMField_17678085390376
// MI455X (gfx1250) — compile-verified
//
#include <hip/hip_runtime.h>
#include <math.h>

// ---------------------------------------------------------------------------
// CAMField on MI455X (gfx1250, wave32).
// Bandwidth need: 2M pts * (8B in + 12B out) = 40MB -> ~1.7us @ 23.3TB/s.
// Compute is tiny (~0.8 GFLOP fp32), so this is VALU/LDS-throughput bound.
// Design: grids cached in LDS channel-innermost (aligned b64 gathers),
// one point per lane for elementwise math, and the 6x6 hidden matmul done
// with chained V_WMMA_F32_16X16X4_F32 on 16-point tiles (2 tiles per wave).
// ---------------------------------------------------------------------------

#define HIDDEN   6
#define GRID_R   16
#define OUT_DIM  3
#define LN_EPS   1e-5f

#define BLOCK  256
#define WAVES  (BLOCK / 32)
#define BLOCKS 1024

typedef __attribute__((ext_vector_type(2))) float v2f;
typedef __attribute__((ext_vector_type(8))) float v8f;

__device__ __forceinline__ float silu_f(float v) {
    return v / (1.0f + __expf(-v));
}

// Bilinear gather of 6 channels from an LDS grid region laid out as
// [(y*16+x)*6 + c].  Corner base offsets are multiples of 24B -> every v2f
// load is 8B-aligned (ds_load_b64).
__device__ __forceinline__ void sample6(const float* __restrict__ base,
                                        int i00, int i01, int i10, int i11,
                                        float wx, float wy, float o[6]) {
    v2f p00a = *(const v2f*)(base + i00);
    v2f p00b = *(const v2f*)(base + i00 + 2);
    v2f p00c = *(const v2f*)(base + i00 + 4);
    v2f p01a = *(const v2f*)(base + i01);
    v2f p01b = *(const v2f*)(base + i01 + 2);
    v2f p01c = *(const v2f*)(base + i01 + 4);
    v2f p10a = *(const v2f*)(base + i10);
    v2f p10b = *(const v2f*)(base + i10 + 2);
    v2f p10c = *(const v2f*)(base + i10 + 4);
    v2f p11a = *(const v2f*)(base + i11);
    v2f p11b = *(const v2f*)(base + i11 + 2);
    v2f p11c = *(const v2f*)(base + i11 + 4);
    float g00[6] = {p00a.x, p00a.y, p00b.x, p00b.y, p00c.x, p00c.y};
    float g01[6] = {p01a.x, p01a.y, p01b.x, p01b.y, p01c.x, p01c.y};
    float g10[6] = {p10a.x, p10a.y, p10b.x, p10b.y, p10c.x, p10c.y};
    float g11[6] = {p11a.x, p11a.y, p11b.x, p11b.y, p11c.x, p11c.y};
#pragma unroll
    for (int c = 0; c < 6; ++c) {
        float top = g00[c] + (g01[c] - g00[c]) * wx;
        float bot = g10[c] + (g11[c] - g10[c]) * wx;
        o[c] = top + (bot - top) * wy;
    }
}

__global__ __launch_bounds__(BLOCK) void camfield_kernel(
    const float* __restrict__ xy,   const float* __restrict__ gamma,
    const float* __restrict__ beta, const float* __restrict__ w_in,
    const float* __restrict__ b_in, const float* __restrict__ w_h,
    const float* __restrict__ b_h,  const float* __restrict__ w_out,
    const float* __restrict__ b_out,const float* __restrict__ ln_w,
    const float* __restrict__ ln_b, float* __restrict__ out, int n)
{
    // LDS: 4 grids of 16*16*6 floats (channel innermost), plus per-wave
    // staging for the WMMA hidden-layer matmul.  ~48KB << 320KB/WGP.
    __shared__ float gridLDS[4 * 1536];         // [layer*2 + {0:gamma,1:beta}]
    __shared__ float stage[WAVES * 32 * 8];     // A-fragment staging
    __shared__ float unpk [WAVES * 2 * 16 * 16];// D-fragment unpack

    const int tid = threadIdx.x;

    // Cooperative transposed copy: src [layer][c][y][x] -> dst [(y*16+x)*6+c]
    for (int i = tid; i < 2 * 6 * 256; i += BLOCK) {
        int l = i / 1536;
        int c = (i / 256) % 6;
        int p = i % 256;
        gridLDS[(l * 2 + 0) * 1536 + p * 6 + c] = gamma[i];
        gridLDS[(l * 2 + 1) * 1536 + p * 6 + c] = beta[i];
    }
    __syncthreads();

    const int lane = tid & 31;
    const int wave = tid >> 5;
    const int n16  = lane & 15;
    const bool hi  = lane >= 16;
    const int k0   = hi ? 2 : 0;

    // ---- loop-invariant parameters (uniform -> SGPRs) ----
    float Wi[6][2], Bi[6], Lw[2][6], Lb[2][6], Wo[3][6], Bo[3];
#pragma unroll
    for (int j = 0; j < 6; ++j) {
        Wi[j][0] = w_in[j * 2 + 0];
        Wi[j][1] = w_in[j * 2 + 1];
        Bi[j]    = b_in[j];
        Lw[0][j] = ln_w[j];       Lw[1][j] = ln_w[6 + j];
        Lb[0][j] = ln_b[j];       Lb[1][j] = ln_b[6 + j];
    }
#pragma unroll
    for (int o = 0; o < 3; ++o) {
        Bo[o] = b_out[o];
#pragma unroll
        for (int j = 0; j < 6; ++j) Wo[o][j] = w_out[o * 6 + j];
    }

    // ---- WMMA B fragments for h @ w_h^T (B[k][n] = w_h[n][k]), K padded ----
    // f32 4x16 B layout: VGPR0/1 hold rows K0/K1 for lanes 0-15, K2/K3 for
    // lanes 16-31 (mirrors the documented 16x4 A layout).
    v2f Blo, Bhi;
    Blo.x = (n16 < 6) ? w_h[n16 * 6 + k0]     : 0.0f;
    Blo.y = (n16 < 6) ? w_h[n16 * 6 + k0 + 1] : 0.0f;
    Bhi.x = (!hi && n16 < 6) ? w_h[n16 * 6 + 4] : 0.0f;  // K=4
    Bhi.y = (!hi && n16 < 6) ? w_h[n16 * 6 + 5] : 0.0f;  // K=5
    const float cbias = (n16 < 6) ? b_h[n16] : 0.0f;
    v8f Cb;
#pragma unroll
    for (int r = 0; r < 8; ++r) Cb[r] = cbias;

    float* st = &stage[wave * 256];
    float* up = &unpk[wave * 512];

    const int stride = (int)(gridDim.x * BLOCK);
    for (int idx = (int)(blockIdx.x * BLOCK) + tid; idx < n; idx += stride) {
        const float x = xy[idx * 2 + 0];
        const float y = xy[idx * 2 + 1];

        // bilinear coords (border clamp, align_corners=True), shared by layers
        float fx = fminf(fmaxf((x + 1.0f) * 0.5f * 15.0f, 0.0f), 15.0f);
        float fy = fminf(fmaxf((y + 1.0f) * 0.5f * 15.0f, 0.0f), 15.0f);
        float x0f = floorf(fx), y0f = floorf(fy);
        float wx = fx - x0f, wy = fy - y0f;
        int x0 = (int)x0f, y0 = (int)y0f;
        int x1 = (x0 < 15) ? x0 + 1 : 15;
        int y1 = (y0 < 15) ? y0 + 1 : 15;
        int i00 = (y0 * 16 + x0) * 6, i01 = (y0 * 16 + x1) * 6;
        int i10 = (y1 * 16 + x0) * 6, i11 = (y1 * 16 + x1) * 6;

        // ---- input layer ----
        float h[6];
#pragma unroll
        for (int j = 0; j < 6; ++j)
            h[j] = silu_f(Wi[j][0] * x + Wi[j][1] * y + Bi[j]);

        // ---- layer 0: FiLM(LayerNorm(h)) ----
        {
            float g[6], bb[6];
            sample6(&gridLDS[0 * 1536], i00, i01, i10, i11, wx, wy, g);
            sample6(&gridLDS[1 * 1536], i00, i01, i10, i11, wx, wy, bb);
            float mu = (h[0]+h[1]+h[2]+h[3]+h[4]+h[5]) * (1.0f/6.0f);
            float var = 0.0f;
#pragma unroll
            for (int j = 0; j < 6; ++j) { float d = h[j]-mu; var += d*d; }
            var *= (1.0f/6.0f);
            float rs = rsqrtf(var + LN_EPS);
#pragma unroll
            for (int j = 0; j < 6; ++j)
                h[j] = g[j] * ((h[j]-mu) * rs * Lw[0][j] + Lb[0][j]) + bb[j];
        }

        // ---- hidden 6x6 matmul via chained V_WMMA_F32_16X16X4_F32 ----
        // Stage per-lane h into wave-private LDS in A-fragment order.
        *(v2f*)&st[lane * 8 + 0] = (v2f){h[0], h[1]};
        *(v2f*)&st[lane * 8 + 2] = (v2f){h[2], h[3]};
        *(v2f*)&st[lane * 8 + 4] = (v2f){h[4], h[5]};
        *(v2f*)&st[lane * 8 + 6] = (v2f){0.0f, 0.0f};   // K=6,7 pad
        // A fragments: tile0 = points 0-15, tile1 = points 16-31.
        v2f a0lo = *(v2f*)&st[n16 * 8 + k0];
        v2f a0hi = *(v2f*)&st[n16 * 8 + 4 + k0];
        v2f a1lo = *(v2f*)&st[(16 + n16) * 8 + k0];
        v2f a1hi = *(v2f*)&st[(16 + n16) * 8 + 4 + k0];

        v8f d0 = __builtin_amdgcn_wmma_f32_16x16x4_f32(
            false, a0lo, false, Blo, (short)0, Cb, false, false);
        d0 = __builtin_amdgcn_wmma_f32_16x16x4_f32(
            false, a0hi, false, Bhi, (short)0, d0, false, false);
        v8f d1 = __builtin_amdgcn_wmma_f32_16x16x4_f32(
            false, a1lo, false, Blo, (short)0, Cb, false, false);
        d1 = __builtin_amdgcn_wmma_f32_16x16x4_f32(
            false, a1hi, false, Bhi, (short)0, d1, false, false);

        // Unpack D (M=point rows, N=channel cols) back to per-lane h.
        if (n16 < 6) {
#pragma unroll
            for (int r = 0; r < 8; ++r) {
                int m = r + (hi ? 8 : 0);
                up[m * 16 + n16]       = d0[r];
                up[256 + m * 16 + n16] = d1[r];
            }
        }
        {
            const float* myrow = &up[(lane >> 4) * 256 + n16 * 16];
            v2f r01 = *(const v2f*)&myrow[0];
            v2f r23 = *(const v2f*)&myrow[2];
            v2f r45 = *(const v2f*)&myrow[4];
            h[0] = silu_f(r01.x); h[1] = silu_f(r01.y);
            h[2] = silu_f(r23.x); h[3] = silu_f(r23.y);
            h[4] = silu_f(r45.x); h[5] = silu_f(r45.y);
        }

        // ---- layer 1: FiLM(LayerNorm(h)) ----
        {
            float g[6], bb[6];
            sample6(&gridLDS[2 * 1536], i00, i01, i10, i11, wx, wy, g);
            sample6(&gridLDS[3 * 1536], i00, i01, i10, i11, wx, wy, bb);
            float mu = (h[0]+h[1]+h[2]+h[3]+h[4]+h[5]) * (1.0f/6.0f);
            float var = 0.0f;
#pragma unroll
            for (int j = 0; j < 6; ++j) { float d = h[j]-mu; var += d*d; }
            var *= (1.0f/6.0f);
            float rs = rsqrtf(var + LN_EPS);
#pragma unroll
            for (int j = 0; j < 6; ++j)
                h[j] = g[j] * ((h[j]-mu) * rs * Lw[1][j] + Lb[1][j]) + bb[j];
        }

        // ---- output projection (3x6, per-lane) ----
#pragma unroll
        for (int o = 0; o < 3; ++o) {
            float acc = Bo[o];
#pragma unroll
            for (int j = 0; j < 6; ++j) acc += Wo[o][j] * h[j];
            out[idx * 3 + o] = acc;
        }
    }
}

extern "C" void kernel_launch(void* const* d_in, const int* in_sizes, int n_in,
                              void* d_out, int out_size, void* d_ws, size_t ws_size,
                              hipStream_t stream) {
    const float* xy    = (const float*)d_in[0];
    const float* gamma = (const float*)d_in[1];
    const float* beta  = (const float*)d_in[2];
    const float* w_in  = (const float*)d_in[3];
    const float* b_in  = (const float*)d_in[4];
    const float* w_h   = (const float*)d_in[5];
    const float* b_h   = (const float*)d_in[6];
    const float* w_out = (const float*)d_in[7];
    const float* b_out = (const float*)d_in[8];
    const float* ln_w  = (const float*)d_in[9];
    const float* ln_b  = (const float*)d_in[10];
    float* out = (float*)d_out;
    int n = in_sizes[0] / 2;   // 2,097,152 points (multiple of 32)

    camfield_kernel<<<BLOCKS, BLOCK, 0, stream>>>(
        xy, gamma, beta, w_in, b_in, w_h, b_h, w_out, b_out, ln_w, ln_b,
        out, n);
}